// ToyCausalTransformer_54176717471830
// MI455X (gfx1250) — compile-verified
//
#include <hip/hip_runtime.h>
#include <math.h>

// ToyCausalTransformer forward on gfx1250 (CDNA5), fp32 end-to-end.
// All GEMM-shaped work uses V_WMMA_F32_16X16X4_F32. Each wave computes a
// 16x64 output strip (4 accumulators share one A fragment per k-step),
// raising arithmetic intensity vs. the naive 16x16/wave version.

typedef float v2f __attribute__((ext_vector_type(2)));
typedef float v8f __attribute__((ext_vector_type(8)));

#define SLEN   2048
#define DMODEL 1024
#define NHEAD  8
#define HDIM   128
#define NLAYER 6

// ---------------------------------------------------------------- embedding
__global__ void embed_kernel(const int* __restrict__ ids,
                             const float* __restrict__ tok,
                             const float* __restrict__ pos,
                             float* __restrict__ out) {
    int s = blockIdx.x;
    int id = ids[s];
    const float* t = tok + (size_t)id * DMODEL;
    const float* p = pos + (size_t)s  * DMODEL;
    float* o = out + (size_t)s * DMODEL;
    for (int d = threadIdx.x; d < DMODEL; d += blockDim.x)
        o[d] = t[d] + p[d];
}

// ---------------------------------------------------------------- layernorm
__global__ void layernorm_kernel(const float* __restrict__ x,
                                 const float* __restrict__ g,
                                 const float* __restrict__ b,
                                 float* __restrict__ y) {
    __shared__ float sh1[256], sh2[256];
    int row = blockIdx.x, tid = threadIdx.x;
    const float* xr = x + (size_t)row * DMODEL;
    float* yr = y + (size_t)row * DMODEL;
    float s = 0.f, s2 = 0.f;
    for (int i = tid; i < DMODEL; i += 256) { float v = xr[i]; s += v; s2 += v * v; }
    sh1[tid] = s; sh2[tid] = s2; __syncthreads();
    for (int off = 128; off > 0; off >>= 1) {
        if (tid < off) { sh1[tid] += sh1[tid + off]; sh2[tid] += sh2[tid + off]; }
        __syncthreads();
    }
    float mean = sh1[0] * (1.0f / DMODEL);
    float var  = sh2[0] * (1.0f / DMODEL) - mean * mean;
    float inv  = rsqrtf(var + 1e-5f);
    for (int i = tid; i < DMODEL; i += 256)
        yr[i] = (xr[i] - mean) * inv * g[i] + b[i];
}

// ------------------------------------------------- generic WMMA fp32 GEMM
// C[M,N] = act(A[M,K] @ W[N,K]^T + bias[N]) (+ residual[M,N])
// grid = (N/64, M/16), block = 32 (one wave -> 16x64 strip, 4 WMMA tiles).
__global__ void gemm_wmma_kernel(const float* __restrict__ A,
                                 const float* __restrict__ W,
                                 const float* __restrict__ bias,
                                 const float* __restrict__ residual,
                                 float* __restrict__ C,
                                 int N, int K, int act) {
    int tn   = blockIdx.x * 64;
    int tm   = blockIdx.y * 16;
    int lane = threadIdx.x;
    int half = lane >> 4;     // 0: K={0,1}, 1: K={2,3} within each 4-step
    int l16  = lane & 15;
    const float* ar = A + (size_t)(tm + l16) * K + 2 * half;
    const float* wr = W + (size_t)(tn + l16) * K + 2 * half;
    const size_t cs = (size_t)16 * K;               // 16 columns of W
    v8f acc0 = {}, acc1 = {}, acc2 = {}, acc3 = {};
    for (int k = 0; k < K; k += 4) {
        __builtin_prefetch(&wr[k + 256], 0, 1);     // stream next W block
        v2f a, b0, b1, b2, b3;
        a.x  = ar[k];          a.y  = ar[k + 1];
        b0.x = wr[k];          b0.y = wr[k + 1];
        b1.x = wr[k + cs];     b1.y = wr[k + cs + 1];
        b2.x = wr[k + 2 * cs]; b2.y = wr[k + 2 * cs + 1];
        b3.x = wr[k + 3 * cs]; b3.y = wr[k + 3 * cs + 1];
        acc0 = __builtin_amdgcn_wmma_f32_16x16x4_f32(false, a, false, b0, (short)0, acc0, false, false);
        acc1 = __builtin_amdgcn_wmma_f32_16x16x4_f32(false, a, false, b1, (short)0, acc1, false, false);
        acc2 = __builtin_amdgcn_wmma_f32_16x16x4_f32(false, a, false, b2, (short)0, acc2, false, false);
        acc3 = __builtin_amdgcn_wmma_f32_16x16x4_f32(false, a, false, b3, (short)0, acc3, false, false);
    }
    v8f accs[4] = {acc0, acc1, acc2, acc3};
#pragma unroll
    for (int t = 0; t < 4; ++t) {
        int col = tn + 16 * t + l16;
        float bv = bias[col];
#pragma unroll
        for (int r = 0; r < 8; ++r) {
            int row = tm + r + 8 * half;
            float v = accs[t][r] + bv;
            if (act == 1) v = 0.5f * v * (1.0f + erff(v * 0.70710678118654752f)); // exact GELU
            if (residual) v += residual[(size_t)row * N + col];
            C[(size_t)row * N + col] = v;
        }
    }
}

// ------------------------------------------- attention scores (pre-softmax)
// scores[h,q,k] = (q . k) / sqrt(HD), causal mask -> -inf. K-dim = HDIM.
// grid = (S/64, S/16, H); each wave -> 16 queries x 64 keys.
__global__ void attn_scores_kernel(const float* __restrict__ qkv,
                                   float* __restrict__ scores) {
    int tn = blockIdx.x * 64;          // key tile base
    int tm = blockIdx.y * 16;          // query tile base
    int h  = blockIdx.z;
    int lane = threadIdx.x, half = lane >> 4, l16 = lane & 15;
    float* out = scores + (size_t)h * SLEN * SLEN;
    if (tn > tm + 15) {                // whole 64-strip above diagonal: masked
#pragma unroll
        for (int t = 0; t < 4; ++t)
#pragma unroll
            for (int r = 0; r < 8; ++r)
                out[(size_t)(tm + r + 8 * half) * SLEN + tn + 16 * t + l16] = -__builtin_inff();
        return;
    }
    const float* qr = qkv + (size_t)(tm + l16) * (3 * DMODEL) + h * HDIM + 2 * half;
    const float* kr = qkv + (size_t)(tn + l16) * (3 * DMODEL) + DMODEL + h * HDIM + 2 * half;
    const size_t ks = (size_t)16 * (3 * DMODEL);   // 16 key rows
    // number of 16-key subtiles that contain any unmasked element (uniform)
    int live = (tm + 15 - tn) / 16 + 1; if (live > 4) live = 4;
    v8f acc0 = {}, acc1 = {}, acc2 = {}, acc3 = {};
    for (int k = 0; k < HDIM; k += 4) {
        v2f a, b;
        a.x = qr[k]; a.y = qr[k + 1];
        b.x = kr[k]; b.y = kr[k + 1];
        acc0 = __builtin_amdgcn_wmma_f32_16x16x4_f32(false, a, false, b, (short)0, acc0, false, false);
        if (live > 1) {
            b.x = kr[k + ks]; b.y = kr[k + ks + 1];
            acc1 = __builtin_amdgcn_wmma_f32_16x16x4_f32(false, a, false, b, (short)0, acc1, false, false);
        }
        if (live > 2) {
            b.x = kr[k + 2 * ks]; b.y = kr[k + 2 * ks + 1];
            acc2 = __builtin_amdgcn_wmma_f32_16x16x4_f32(false, a, false, b, (short)0, acc2, false, false);
        }
        if (live > 3) {
            b.x = kr[k + 3 * ks]; b.y = kr[k + 3 * ks + 1];
            acc3 = __builtin_amdgcn_wmma_f32_16x16x4_f32(false, a, false, b, (short)0, acc3, false, false);
        }
    }
    const float scale = 0.08838834764831845f;   // 1/sqrt(128)
    v8f accs[4] = {acc0, acc1, acc2, acc3};
#pragma unroll
    for (int t = 0; t < 4; ++t) {
#pragma unroll
        for (int r = 0; r < 8; ++r) {
            int row = tm + r + 8 * half, col = tn + 16 * t + l16;
            float v = accs[t][r] * scale;
            if (col > row) v = -__builtin_inff();
            out[(size_t)row * SLEN + col] = v;
        }
    }
}

// ------------------------------------------------------- row softmax in place
__global__ void softmax_kernel(float* __restrict__ a) {
    __shared__ float sh[256];
    int row = blockIdx.x, tid = threadIdx.x;
    float* r = a + (size_t)row * SLEN;
    float m = -__builtin_inff();
    for (int i = tid; i < SLEN; i += 256) m = fmaxf(m, r[i]);
    sh[tid] = m; __syncthreads();
    for (int off = 128; off > 0; off >>= 1) {
        if (tid < off) sh[tid] = fmaxf(sh[tid], sh[tid + off]);
        __syncthreads();
    }
    m = sh[0]; __syncthreads();
    float s = 0.f;
    for (int i = tid; i < SLEN; i += 256) {
        float e = expf(r[i] - m);   // exp(-inf - m) = 0 handles the mask
        r[i] = e; s += e;
    }
    sh[tid] = s; __syncthreads();
    for (int off = 128; off > 0; off >>= 1) {
        if (tid < off) sh[tid] += sh[tid + off];
        __syncthreads();
    }
    float inv = 1.0f / sh[0];
    for (int i = tid; i < SLEN; i += 256) r[i] *= inv;
}

// ------------------------------------------------- ctx = attn @ V per head
// ctx[q, h*HD + n] = sum_k attn[h,q,k] * v[k, h*HD + n]; K bounded by causality.
// grid = (HDIM/64 = 2, S/16, H); each wave -> 16 queries x 64 head-dims.
__global__ void attn_ctx_kernel(const float* __restrict__ attn,
                                const float* __restrict__ qkv,
                                float* __restrict__ ctx) {
    int tn = blockIdx.x * 64;          // within HDIM
    int tm = blockIdx.y * 16;          // query tile
    int h  = blockIdx.z;
    int lane = threadIdx.x, half = lane >> 4, l16 = lane & 15;
    const float* ar = attn + (size_t)h * SLEN * SLEN + (size_t)(tm + l16) * SLEN + 2 * half;
    const float* vb = qkv + 2 * DMODEL + h * HDIM + tn + l16;   // v[k][n] at vb[k*3D]
    v8f acc0 = {}, acc1 = {}, acc2 = {}, acc3 = {};
    int kmax = tm + 16;                // attn[q,k]==0 for k>q, max q = tm+15
    for (int k = 0; k < kmax; k += 4) {
        size_t r0 = (size_t)(k + 2 * half) * (3 * DMODEL);
        size_t r1 = (size_t)(k + 2 * half + 1) * (3 * DMODEL);
        v2f a, b;
        a.x = ar[k]; a.y = ar[k + 1];
        b.x = vb[r0];      b.y = vb[r1];
        acc0 = __builtin_amdgcn_wmma_f32_16x16x4_f32(false, a, false, b, (short)0, acc0, false, false);
        b.x = vb[r0 + 16]; b.y = vb[r1 + 16];
        acc1 = __builtin_amdgcn_wmma_f32_16x16x4_f32(false, a, false, b, (short)0, acc1, false, false);
        b.x = vb[r0 + 32]; b.y = vb[r1 + 32];
        acc2 = __builtin_amdgcn_wmma_f32_16x16x4_f32(false, a, false, b, (short)0, acc2, false, false);
        b.x = vb[r0 + 48]; b.y = vb[r1 + 48];
        acc3 = __builtin_amdgcn_wmma_f32_16x16x4_f32(false, a, false, b, (short)0, acc3, false, false);
    }
    v8f accs[4] = {acc0, acc1, acc2, acc3};
#pragma unroll
    for (int t = 0; t < 4; ++t)
#pragma unroll
        for (int r = 0; r < 8; ++r) {
            int row = tm + r + 8 * half;
            ctx[(size_t)row * DMODEL + h * HDIM + tn + 16 * t + l16] = accs[t][r];
        }
}

// ------------------------------------------------------------------ driver
extern "C" void kernel_launch(void* const* d_in, const int* in_sizes, int n_in,
                              void* d_out, int out_size, void* d_ws, size_t ws_size,
                              hipStream_t stream) {
    const int S = SLEN, D = DMODEL, L = NLAYER, H = NHEAD;

    const int*   token_ids = (const int*)  d_in[0];
    const float* tok_emb   = (const float*)d_in[1];
    const float* pos_emb   = (const float*)d_in[2];
    const float* ln1_g     = (const float*)d_in[3];
    const float* ln1_b     = (const float*)d_in[4];
    const float* in_w      = (const float*)d_in[5];
    const float* in_b      = (const float*)d_in[6];
    const float* out_w     = (const float*)d_in[7];
    const float* out_b     = (const float*)d_in[8];
    const float* ln2_g     = (const float*)d_in[9];
    const float* ln2_b     = (const float*)d_in[10];
    const float* w1        = (const float*)d_in[11];
    const float* b1        = (const float*)d_in[12];
    const float* w2        = (const float*)d_in[13];
    const float* b2        = (const float*)d_in[14];

    float* hiddens = (float*)d_out;                                // [L+1,S,D]
    float* attns   = hiddens + (size_t)(L + 1) * S * D;            // [L,H,S,S]

    char*  ws    = (char*)d_ws;
    float* h_buf = (float*)(ws);                                   // [S,D]   8 MiB
    float* qkv   = (float*)(ws + (size_t)S * D * 4);               // [S,3D] 24 MiB
    float* ctx   = (float*)(ws + (size_t)S * D * 4 * 4);           // [S,D]   8 MiB
    float* xmid  = (float*)(ws + (size_t)S * D * 4 * 5);           // [S,D]   8 MiB
    float* tbuf  = (float*)(ws + (size_t)S * D * 4 * 6);           // [S,4D] 32 MiB

    embed_kernel<<<S, 256, 0, stream>>>(token_ids, tok_emb, pos_emb, hiddens);

    for (int l = 0; l < L; ++l) {
        float* x   = hiddens + (size_t)l * S * D;
        float* xn  = hiddens + (size_t)(l + 1) * S * D;
        float* att = attns   + (size_t)l * H * S * S;

        // --- attention block ---
        layernorm_kernel<<<S, 256, 0, stream>>>(x, ln1_g + (size_t)l * D,
                                                ln1_b + (size_t)l * D, h_buf);
        gemm_wmma_kernel<<<dim3(3 * D / 64, S / 16), 32, 0, stream>>>(
            h_buf, in_w + (size_t)l * 3 * D * D, in_b + (size_t)l * 3 * D,
            nullptr, qkv, 3 * D, D, 0);
        attn_scores_kernel<<<dim3(S / 64, S / 16, H), 32, 0, stream>>>(qkv, att);
        softmax_kernel<<<H * S, 256, 0, stream>>>(att);
        attn_ctx_kernel<<<dim3(HDIM / 64, S / 16, H), 32, 0, stream>>>(att, qkv, ctx);
        gemm_wmma_kernel<<<dim3(D / 64, S / 16), 32, 0, stream>>>(
            ctx, out_w + (size_t)l * D * D, out_b + (size_t)l * D,
            /*residual=*/x, xmid, D, D, 0);

        // --- MLP block ---
        layernorm_kernel<<<S, 256, 0, stream>>>(xmid, ln2_g + (size_t)l * D,
                                                ln2_b + (size_t)l * D, h_buf);
        gemm_wmma_kernel<<<dim3(4 * D / 64, S / 16), 32, 0, stream>>>(
            h_buf, w1 + (size_t)l * 4 * D * D, b1 + (size_t)l * 4 * D,
            nullptr, tbuf, 4 * D, D, /*gelu=*/1);
        gemm_wmma_kernel<<<dim3(D / 64, S / 16), 32, 0, stream>>>(
            tbuf, w2 + (size_t)l * D * 4 * D, b2 + (size_t)l * D,
            /*residual=*/xmid, xn, D, 4 * D, 0);
    }
    (void)in_sizes; (void)n_in; (void)out_size; (void)ws_size;
}